// cheerla_et_al_criterion_4389456576887
// MI455X (gfx1250) — compile-verified
//
#include <hip/hip_runtime.h>
#include <hip/hip_bf16.h>

typedef __attribute__((ext_vector_type(2))) float v2f;
typedef __attribute__((ext_vector_type(8))) float v8f;

#define NPTS 4096
#define DIM  256
#define NMOD 4
#define TILE 128
#define KC   32
#define LSTR (KC + 4)   // padded LDS row stride (floats); 36*4=144B keeps 16B alignment

// ---------------------------------------------------------------------------
// Preprocess: per row (mod*N + i): norm, normalized embedding, self-sim, valid mask
// ---------------------------------------------------------------------------
__global__ __launch_bounds__(256) void prep_kernel(const float* __restrict__ enc,
                                                   float* __restrict__ xn,
                                                   float* __restrict__ selfsim,
                                                   float* __restrict__ vmask) {
  __shared__ float red[256];
  __shared__ int   eq[256];
  const int tid = threadIdx.x;
  const int row = blockIdx.x;                 // row in [0, NMOD*NPTS)
  const float* x = enc + (size_t)row * DIM;
  float v  = x[tid];
  float x0 = x[0];
  red[tid] = v * v;
  eq[tid]  = (v == x0) ? 1 : 0;
  __syncthreads();
  #pragma unroll
  for (int s = 128; s > 0; s >>= 1) {
    if (tid < s) { red[tid] += red[tid + s]; eq[tid] &= eq[tid + s]; }
    __syncthreads();
  }
  float sumsq = red[0];
  int   miss  = eq[0];
  float nrm   = sqrtf(sumsq);
  float scale = 1.0f / fmaxf(nrm, 1e-8f);
  xn[(size_t)row * DIM + tid] = v * scale;
  if (tid == 0) {
    selfsim[row] = sumsq * scale * scale;     // cos(x,x) (==1 unless degenerate)
    vmask[row]   = miss ? 0.0f : 1.0f;
  }
}

// ---------------------------------------------------------------------------
// Cox partial likelihood: per-block partial sums of ev*(lh - log(denom)) and ev
// ---------------------------------------------------------------------------
__global__ __launch_bounds__(256) void cox_kernel(const float* __restrict__ lh,
                                                  const float* __restrict__ tme,
                                                  const int*   __restrict__ ev,
                                                  float* __restrict__ coxNum,
                                                  float* __restrict__ coxDen) {
  __shared__ float ts[256];
  __shared__ float es[256];
  __shared__ float red[256];
  const int tid = threadIdx.x;
  const int i   = blockIdx.x * 256 + tid;
  const float ti  = tme[i];
  const float lhi = lh[i];
  float denom = 0.0f;
  for (int c = 0; c < NPTS; c += 256) {
    __syncthreads();
    ts[tid] = tme[c + tid];
    es[tid] = expf(lh[c + tid]);
    __syncthreads();
    #pragma unroll 8
    for (int j = 0; j < 256; ++j)
      denom += (ts[j] >= ti) ? es[j] : 0.0f;
  }
  const float evi = (float)ev[i];
  float num = evi * (lhi - logf(denom));
  __syncthreads();
  red[tid] = num; __syncthreads();
  #pragma unroll
  for (int s = 128; s > 0; s >>= 1) { if (tid < s) red[tid] += red[tid + s]; __syncthreads(); }
  if (tid == 0) coxNum[blockIdx.x] = red[0];
  __syncthreads();
  red[tid] = evi; __syncthreads();
  #pragma unroll
  for (int s = 128; s > 0; s >>= 1) { if (tid < s) red[tid] += red[tid + s]; __syncthreads(); }
  if (tid == 0) coxDen[blockIdx.x] = red[0];
}

// ---------------------------------------------------------------------------
// Similarity loss: 128x128 pair tile per block, WMMA f32 16x16x4 over K=256,
// 4 modalities folded into one persistent pair accumulator, hinge + reduce.
// 8 waves; wave w owns rows [wrow*32, +32) x cols [wcol*64, +64) of the tile.
// ---------------------------------------------------------------------------
__global__ __launch_bounds__(256) void sim_kernel(const float* __restrict__ xn,
                                                  const float* __restrict__ selfsim,
                                                  const float* __restrict__ vmask,
                                                  const float* __restrict__ Mp,
                                                  float* __restrict__ simPartials) {
  __shared__ float ldsA[TILE * LSTR];
  __shared__ float ldsB[TILE * LSTR];

  const int tid  = threadIdx.x;
  const int wave = tid >> 5;
  const int lane = tid & 31;
  const int half = lane >> 4;      // lanes 16-31 handle K+2..K+3 / rows M+8
  const int l16  = lane & 15;
  const int wrow = wave >> 1;      // 0..3 : 32-row band
  const int wcol = wave & 1;       // 0..1 : 64-col band
  const int ti0  = blockIdx.y * TILE;
  const int tj0  = blockIdx.x * TILE;

  const v8f vzero = {0.f, 0.f, 0.f, 0.f, 0.f, 0.f, 0.f, 0.f};

  v8f pacc[2][4];
  #pragma unroll
  for (int si = 0; si < 2; ++si)
    #pragma unroll
    for (int sj = 0; sj < 4; ++sj) pacc[si][sj] = vzero;

  for (int mod = 0; mod < NMOD; ++mod) {
    const float* Xm = xn + (size_t)mod * NPTS * DIM;

    v8f S[2][4];
    #pragma unroll
    for (int si = 0; si < 2; ++si)
      #pragma unroll
      for (int sj = 0; sj < 4; ++sj) S[si][sj] = vzero;

    for (int kc = 0; kc < DIM; kc += KC) {
      __syncthreads();
      // cooperative stage of A rows [ti0..+127] and B rows [tj0..+127], cols [kc..+31]
      #pragma unroll
      for (int p = 0; p < 4; ++p) {
        int f   = tid + p * 256;       // float4 slot, 1024 total (128 rows * 8 f4)
        int row = f >> 3;
        int c4  = f & 7;
        const float4* ga = (const float4*)(Xm + (size_t)(ti0 + row) * DIM + kc) + c4;
        const float4* gb = (const float4*)(Xm + (size_t)(tj0 + row) * DIM + kc) + c4;
        *(float4*)&ldsA[row * LSTR + c4 * 4] = *ga;
        *(float4*)&ldsB[row * LSTR + c4 * 4] = *gb;
      }
      __syncthreads();

      #pragma unroll
      for (int kk = 0; kk < KC; kk += 4) {
        const int col = kk + 2 * half;       // lanes 0-15: K,K+1 ; 16-31: K+2,K+3
        v2f a[2], b[4];
        #pragma unroll
        for (int si = 0; si < 2; ++si)
          a[si] = *(const v2f*)&ldsA[(wrow * 32 + si * 16 + l16) * LSTR + col];
        #pragma unroll
        for (int sj = 0; sj < 4; ++sj)
          b[sj] = *(const v2f*)&ldsB[(wcol * 64 + sj * 16 + l16) * LSTR + col];
        #pragma unroll
        for (int si = 0; si < 2; ++si)
          #pragma unroll
          for (int sj = 0; sj < 4; ++sj)
            S[si][sj] = __builtin_amdgcn_wmma_f32_16x16x4_f32(
                false, a[si], false, b[sj], (short)0, S[si][sj], false, false);
      }
    }

    // fold modality into pair accumulator: pacc += vm_i*vm_j*(S - selfsim_i)
    float vmA[2][8], ssA[2][8];
    #pragma unroll
    for (int si = 0; si < 2; ++si)
      #pragma unroll
      for (int r = 0; r < 8; ++r) {
        int i = ti0 + wrow * 32 + si * 16 + r + 8 * half;   // C layout: M = r + 8*half
        vmA[si][r] = vmask[mod * NPTS + i];
        ssA[si][r] = selfsim[mod * NPTS + i];
      }
    #pragma unroll
    for (int sj = 0; sj < 4; ++sj) {
      int j = tj0 + wcol * 64 + sj * 16 + l16;              // C layout: N = lane&15
      float vmB = vmask[mod * NPTS + j];
      #pragma unroll
      for (int si = 0; si < 2; ++si)
        #pragma unroll
        for (int r = 0; r < 8; ++r)
          pacc[si][sj][r] += vmA[si][r] * vmB * (S[si][sj][r] - ssA[si][r]);
    }
  }

  // hinge + off-diagonal mask, local sum
  const float Mv = Mp[0];
  float local = 0.0f;
  #pragma unroll
  for (int si = 0; si < 2; ++si)
    #pragma unroll
    for (int sj = 0; sj < 4; ++sj)
      #pragma unroll
      for (int r = 0; r < 8; ++r) {
        int i = ti0 + wrow * 32 + si * 16 + r + 8 * half;
        int j = tj0 + wcol * 64 + sj * 16 + l16;
        if (i != j) {
          float h = Mv - pacc[si][sj][r];
          local += (h > 0.0f) ? h : 0.0f;
        }
      }

  __syncthreads();                      // LDS tile reads done; reuse ldsA for reduction
  ldsA[tid] = local;
  __syncthreads();
  #pragma unroll
  for (int s = 128; s > 0; s >>= 1) { if (tid < s) ldsA[tid] += ldsA[tid + s]; __syncthreads(); }
  if (tid == 0) simPartials[blockIdx.y * gridDim.x + blockIdx.x] = ldsA[0];
}

// ---------------------------------------------------------------------------
// Deterministic final combine: sum partials, out = -coxNum/coxDen + simSum
// ws layout: [0..1024) sim partials, [1024..1040) coxNum, [1040..1056) coxDen
// ---------------------------------------------------------------------------
__global__ __launch_bounds__(256) void final_kernel(const float* __restrict__ ws,
                                                    float* __restrict__ out) {
  __shared__ float red[256];
  const int tid = threadIdx.x;

  float s = 0.0f;
  #pragma unroll
  for (int p = 0; p < 4; ++p) s += ws[tid + p * 256];
  red[tid] = s; __syncthreads();
  #pragma unroll
  for (int k = 128; k > 0; k >>= 1) { if (tid < k) red[tid] += red[tid + k]; __syncthreads(); }
  float simSum = red[0];
  __syncthreads();

  red[tid] = (tid < 16) ? ws[1024 + tid] : 0.0f; __syncthreads();
  #pragma unroll
  for (int k = 128; k > 0; k >>= 1) { if (tid < k) red[tid] += red[tid + k]; __syncthreads(); }
  float num = red[0];
  __syncthreads();

  red[tid] = (tid < 16) ? ws[1040 + tid] : 0.0f; __syncthreads();
  #pragma unroll
  for (int k = 128; k > 0; k >>= 1) { if (tid < k) red[tid] += red[tid + k]; __syncthreads(); }
  float den = red[0];

  if (tid == 0) out[0] = -num / den + simSum;
}

// ---------------------------------------------------------------------------
extern "C" void kernel_launch(void* const* d_in, const int* in_sizes, int n_in,
                              void* d_out, int out_size, void* d_ws, size_t ws_size,
                              hipStream_t stream) {
  const float* lh   = (const float*)d_in[0];   // [N]
  const float* enc  = (const float*)d_in[1];   // [m,N,D]
  const float* tme  = (const float*)d_in[2];   // [N]
  const int*   ev   = (const int*)d_in[3];     // [N]
  const float* Mp   = (const float*)d_in[4];   // scalar

  float* ws          = (float*)d_ws;
  float* simPartials = ws;                     // 1024
  float* coxNum      = ws + 1024;              // 16
  float* coxDen      = ws + 1040;              // 16
  float* xn          = ws + 1280;              // m*N*D floats (16B aligned)
  float* selfsim     = xn + (size_t)NMOD * NPTS * DIM;  // m*N
  float* vmask       = selfsim + NMOD * NPTS;           // m*N

  prep_kernel<<<NMOD * NPTS, 256, 0, stream>>>(enc, xn, selfsim, vmask);
  cox_kernel<<<NPTS / 256, 256, 0, stream>>>(lh, tme, ev, coxNum, coxDen);
  dim3 grid(NPTS / TILE, NPTS / TILE);
  sim_kernel<<<grid, 256, 0, stream>>>(xn, selfsim, vmask, Mp, simPartials);
  final_kernel<<<1, 256, 0, stream>>>(ws, (float*)d_out);
}